// HeterogeneousRecommenderGNN_10857677324737
// MI455X (gfx1250) — compile-verified
//
#include <hip/hip_runtime.h>

typedef __attribute__((ext_vector_type(2))) float v2f;
typedef __attribute__((ext_vector_type(8))) float v8f;

#define NU   100000
#define NA   20000
#define FU   128
#define FA   384
#define HDIM 128
#define ODIM 64
#define NEFF 1600000
#define NERV 2000000

// ---------------------------------------------------------------------------
// One WMMA GEMM term over a full 16-row stripe:
//   C[nt] += rowscale(A[16 rows, K]) * W[K, nt*16 .. nt*16+15]   for nt in [0,NT)
// A fragment (f32 16x4): lanes 0-15 hold K={0,1}, lanes 16-31 hold K={2,3},
// row = lane&15. Loaded ONCE per k-step and reused for all NT column tiles;
// the NT WMMAs hit independent accumulators -> no D->A RAW hazard, they can
// issue back-to-back into the matrix pipe.
// Degree normalization (1/max(cnt,1)) folded into the A-fragment load.
// ---------------------------------------------------------------------------
template <int NT>
__device__ __forceinline__ void wmma_term(v8f (&c)[NT],
                                          const float* __restrict__ A, int lda,
                                          const float* __restrict__ cnt,
                                          const float* __restrict__ W,
                                          int rowBase, int K, int lane) {
  const int N    = NT * 16;
  const int m    = rowBase + (lane & 15);
  const int koff = (lane >> 4) << 1;              // 0 for lanes 0-15, 2 for 16-31
  float scale = 1.0f;
  if (cnt) scale = 1.0f / fmaxf(cnt[m], 1.0f);    // segment-mean divisor
  const float* ap = A + (size_t)m * lda + koff;
  const float* wp = W + (size_t)koff * N + (lane & 15);
  for (int k = 0; k < K; k += 4) {
    v2f a;
    a.x = ap[k]     * scale;
    a.y = ap[k + 1] * scale;
    const float* wk0 = wp + (size_t)k * N;
    const float* wk1 = wk0 + N;
#pragma unroll
    for (int nt = 0; nt < NT; ++nt) {
      v2f b;
      b.x = wk0[nt * 16];
      b.y = wk1[nt * 16];
      c[nt] = __builtin_amdgcn_wmma_f32_16x16x4_f32(false, a, false, b,
                                                    (short)0, c[nt],
                                                    false, false);
    }
  }
}

// ---------------------------------------------------------------------------
// Fused SAGE/encoder GEMM (N = NT*16):
//   out = relu( A0/deg0 @ W0 + b0 [+ A1 @ W1] [+ A2/deg2 @ W2 + b2] [+ A3 @ W3] )
// One wave = one 16-row stripe x all NT column tiles. M multiple of 16.
// ---------------------------------------------------------------------------
template <int NT>
__global__ __launch_bounds__(256) void fused_sage_gemm(
    const float* __restrict__ A0, const float* __restrict__ cnt0,
    const float* __restrict__ W0, const float* __restrict__ bias0,
    const float* __restrict__ A1, const float* __restrict__ W1,
    const float* __restrict__ A2, const float* __restrict__ cnt2,
    const float* __restrict__ W2, const float* __restrict__ bias2,
    const float* __restrict__ A3, const float* __restrict__ W3,
    float* __restrict__ out, int M, int K) {
  const int N      = NT * 16;
  const int lane   = threadIdx.x & 31;
  const int wave   = threadIdx.x >> 5;
  const int stripe = blockIdx.x * (blockDim.x >> 5) + wave;
  if (stripe >= (M >> 4)) return;                  // uniform per wave
  const int rowBase = stripe << 4;

  v8f c[NT];
#pragma unroll
  for (int nt = 0; nt < NT; ++nt) c[nt] = (v8f){};

  wmma_term<NT>(c, A0, K, cnt0, W0, rowBase, K, lane);
  if (A1) wmma_term<NT>(c, A1, K, nullptr, W1, rowBase, K, lane);
  if (A2) wmma_term<NT>(c, A2, K, cnt2,   W2, rowBase, K, lane);
  if (A3) wmma_term<NT>(c, A3, K, nullptr, W3, rowBase, K, lane);

  // C/D layout: VGPR v -> row v (lanes 0-15) or row v+8 (lanes 16-31)
  const int rTop = rowBase + ((lane >> 4) << 3);
#pragma unroll
  for (int nt = 0; nt < NT; ++nt) {
    const int col = nt * 16 + (lane & 15);
    float bsum = bias0 ? bias0[col] : 0.0f;
    if (bias2) bsum += bias2[col];
#pragma unroll
    for (int v = 0; v < 8; ++v) {
      float x = c[nt][v] + bsum;
      out[(size_t)(rTop + v) * N + col] = fmaxf(x, 0.0f);
    }
  }
}

// ---------------------------------------------------------------------------
// Edge scatter-add: one wave per edge, float4 gather, 4 f32 atomics per lane.
// Feature width fixed at HDIM=128 (all scatter inputs are H=128).
// ---------------------------------------------------------------------------
__global__ __launch_bounds__(256) void scatter_add_kernel(
    const float* __restrict__ x, const int* __restrict__ src,
    const int* __restrict__ dst, float* __restrict__ agg, int E) {
  const long long t = (long long)blockIdx.x * blockDim.x + threadIdx.x;
  const int e    = (int)(t >> 5);
  const int lane = (int)(t & 31);
  if (e >= E) return;
  const int s = src[e];
  const int d = dst[e];
  const float4 v = ((const float4*)(x + (size_t)s * HDIM))[lane];
  float* ag = agg + (size_t)d * HDIM + lane * 4;
  atomicAdd(ag + 0, v.x);
  atomicAdd(ag + 1, v.y);
  atomicAdd(ag + 2, v.z);
  atomicAdd(ag + 3, v.w);
}

__global__ __launch_bounds__(256) void count_deg_kernel(
    const int* __restrict__ dst, float* __restrict__ cnt, int E) {
  const int e = blockIdx.x * blockDim.x + threadIdx.x;
  if (e < E) atomicAdd(&cnt[dst[e]], 1.0f);
}

// ---------------------------------------------------------------------------
extern "C" void kernel_launch(void* const* d_in, const int* in_sizes, int n_in,
                              void* d_out, int out_size, void* d_ws, size_t ws_size,
                              hipStream_t stream) {
  (void)in_sizes; (void)n_in; (void)out_size; (void)ws_size;

  const float* x_user = (const float*)d_in[0];
  const float* x_app  = (const float*)d_in[1];
  const int* ff_src  = (const int*)d_in[2];
  const int* ff_dst  = (const int*)d_in[3];
  const int* rev_src = (const int*)d_in[4];
  const int* rev_dst = (const int*)d_in[5];
  const int* rb_src  = (const int*)d_in[6];
  const int* rb_dst  = (const int*)d_in[7];
  const float* Wu = (const float*)d_in[8];
  const float* bu = (const float*)d_in[9];
  const float* Wa = (const float*)d_in[10];
  const float* ba = (const float*)d_in[11];
  const float* l1_ff_Wl  = (const float*)d_in[12];
  const float* l1_ff_bl  = (const float*)d_in[13];
  const float* l1_ff_Wr  = (const float*)d_in[14];
  const float* l1_rev_Wl = (const float*)d_in[15];
  const float* l1_rev_bl = (const float*)d_in[16];
  const float* l1_rev_Wr = (const float*)d_in[17];
  const float* l1_rb_Wl  = (const float*)d_in[18];
  const float* l1_rb_bl  = (const float*)d_in[19];
  const float* l1_rb_Wr  = (const float*)d_in[20];
  const float* l2_ff_Wl  = (const float*)d_in[21];
  const float* l2_ff_bl  = (const float*)d_in[22];
  const float* l2_ff_Wr  = (const float*)d_in[23];
  const float* l2_rev_Wl = (const float*)d_in[24];
  const float* l2_rev_bl = (const float*)d_in[25];
  const float* l2_rev_Wr = (const float*)d_in[26];
  const float* l2_rb_Wl  = (const float*)d_in[27];
  const float* l2_rb_bl  = (const float*)d_in[28];
  const float* l2_rb_Wr  = (const float*)d_in[29];

  // ---- workspace carve-out (≈236 MB) -------------------------------------
  float* ws = (float*)d_ws;
  float* u      = ws; ws += (size_t)NU * HDIM;
  float* u1     = ws; ws += (size_t)NU * HDIM;
  float* a      = ws; ws += (size_t)NA * HDIM;
  float* a1     = ws; ws += (size_t)NA * HDIM;
  float* agg_ff = ws; ws += (size_t)NU * HDIM;   // agg buffers contiguous
  float* agg_rb = ws; ws += (size_t)NU * HDIM;
  float* agg_rv = ws; ws += (size_t)NA * HDIM;
  float* cnt_ff = ws; ws += NU;                  // counts contiguous
  float* cnt_rb = ws; ws += NU;
  float* cnt_rv = ws; ws += NA;

  const size_t aggBytes = (size_t)(2 * NU + NA) * HDIM * sizeof(float);
  const size_t cntBytes = (size_t)(2 * NU + NA) * sizeof(float);

  const int gU  = ((NU / 16) + 7) / 8;           // 16-row stripes, 8 waves/block
  const int gA  = ((NA / 16) + 7) / 8;
  const int sgFF = (int)(((long long)NEFF * 32 + 255) / 256);
  const int sgRV = (int)(((long long)NERV * 32 + 255) / 256);

  // ---- degrees (shared by both layers) -----------------------------------
  hipMemsetAsync(cnt_ff, 0, cntBytes, stream);
  count_deg_kernel<<<(NEFF + 255) / 256, 256, 0, stream>>>(ff_dst, cnt_ff, NEFF);
  count_deg_kernel<<<(NERV + 255) / 256, 256, 0, stream>>>(rb_dst, cnt_rb, NERV);
  count_deg_kernel<<<(NERV + 255) / 256, 256, 0, stream>>>(rev_dst, cnt_rv, NERV);

  // ---- encoders -----------------------------------------------------------
  fused_sage_gemm<HDIM / 16><<<gU, 256, 0, stream>>>(x_user, nullptr, Wu, bu,
      nullptr, nullptr, nullptr, nullptr, nullptr, nullptr, nullptr, nullptr,
      u, NU, FU);
  fused_sage_gemm<HDIM / 16><<<gA, 256, 0, stream>>>(x_app, nullptr, Wa, ba,
      nullptr, nullptr, nullptr, nullptr, nullptr, nullptr, nullptr, nullptr,
      a, NA, FA);

  // ---- layer 1 ------------------------------------------------------------
  hipMemsetAsync(agg_ff, 0, aggBytes, stream);
  scatter_add_kernel<<<sgFF, 256, 0, stream>>>(u, ff_src, ff_dst, agg_ff, NEFF);
  scatter_add_kernel<<<sgRV, 256, 0, stream>>>(a, rb_src, rb_dst, agg_rb, NERV);
  scatter_add_kernel<<<sgRV, 256, 0, stream>>>(u, rev_src, rev_dst, agg_rv, NERV);

  fused_sage_gemm<HDIM / 16><<<gU, 256, 0, stream>>>(agg_ff, cnt_ff, l1_ff_Wl, l1_ff_bl,
      u, l1_ff_Wr, agg_rb, cnt_rb, l1_rb_Wl, l1_rb_bl, u, l1_rb_Wr,
      u1, NU, HDIM);
  fused_sage_gemm<HDIM / 16><<<gA, 256, 0, stream>>>(agg_rv, cnt_rv, l1_rev_Wl, l1_rev_bl,
      a, l1_rev_Wr, nullptr, nullptr, nullptr, nullptr, nullptr, nullptr,
      a1, NA, HDIM);

  // ---- layer 2 ------------------------------------------------------------
  hipMemsetAsync(agg_ff, 0, aggBytes, stream);
  scatter_add_kernel<<<sgFF, 256, 0, stream>>>(u1, ff_src, ff_dst, agg_ff, NEFF);
  scatter_add_kernel<<<sgRV, 256, 0, stream>>>(a1, rb_src, rb_dst, agg_rb, NERV);
  scatter_add_kernel<<<sgRV, 256, 0, stream>>>(u1, rev_src, rev_dst, agg_rv, NERV);

  float* u2 = (float*)d_out;
  float* a2 = (float*)d_out + (size_t)NU * ODIM;
  fused_sage_gemm<ODIM / 16><<<gU, 256, 0, stream>>>(agg_ff, cnt_ff, l2_ff_Wl, l2_ff_bl,
      u1, l2_ff_Wr, agg_rb, cnt_rb, l2_rb_Wl, l2_rb_bl, u1, l2_rb_Wr,
      u2, NU, HDIM);
  fused_sage_gemm<ODIM / 16><<<gA, 256, 0, stream>>>(agg_rv, cnt_rv, l2_rev_Wl, l2_rev_bl,
      a1, l2_rev_Wr, nullptr, nullptr, nullptr, nullptr, nullptr, nullptr,
      a2, NA, HDIM);
}